// GNN_layer_44495861186881
// MI455X (gfx1250) — compile-verified
//
#include <hip/hip_runtime.h>
#include <hip/hip_bf16.h>

// ---------------------------------------------------------------------------
// GNN layer, fused for MI455X (gfx1250, wave32).
//   edges   = leaky(z_mk @ Wedge^T + bcast(z_m @ Wm^T) + bcast(z_k @ Wk^T))
//   msg_nk  = mean_m(edges); msg_nm = mean_k(edges)
//   z_m_upd = leaky(z_m @ Wself_m^T + msg_nm @ Wneigh_m^T)   (z_k analogous)
// All GEMMs use V_WMMA_F32_16X16X4_F32 chained over K=64.
// Edge-kernel A-tile staging uses GLOBAL_LOAD_ASYNC_TO_LDS_B128 (ASYNCcnt).
// ---------------------------------------------------------------------------

typedef __attribute__((ext_vector_type(2))) float v2f;
typedef __attribute__((ext_vector_type(4))) float v4f;
typedef __attribute__((ext_vector_type(8))) float v8f;

#define MDIM 64
#define KDIM 64
#define FDIM 64
#define BATCH 128
#define LDA 68          // padded LDS row stride (dwords): bank-conflict free, 16B aligned

#define N_MK   (BATCH * MDIM * KDIM * FDIM)   // 33554432
#define N_NODE (BATCH * MDIM * FDIM)          // 524288

__device__ __forceinline__ float leaky(float x) {
    return x >= 0.0f ? x : 0.01f * x;
}

__device__ __forceinline__ void atomic_add_f32(float* p, float v) {
    // lowers to global_atomic_add_f32 / ds_add_f32 (no CAS loop)
    unsafeAtomicAdd(p, v);
}

// Async global -> LDS copy of 16 bytes per lane. The VDST operand is the LDS
// byte address: per ISA 10.2, the low 32 bits of a generic shared pointer ARE
// the LDS address (aperture maps addr[31:0] directly), so truncation suffices.
__device__ __forceinline__ void async_copy_b128(float* lds_dst, const float* gsrc) {
    uint32_t loff = (uint32_t)(uintptr_t)lds_dst;
    asm volatile("global_load_async_to_lds_b128 %0, %1, off"
                 :: "v"(loff), "v"(gsrc) : "memory");
}

__device__ __forceinline__ void wait_async_all() {
    asm volatile("s_wait_asynccnt 0x0" ::: "memory");
}

// One WMMA K-step: acc += A(16x4) * B(4x16), f32.
__device__ __forceinline__ v8f wmma_step(v2f a, v2f b, v8f acc) {
    return __builtin_amdgcn_wmma_f32_16x16x4_f32(
        /*neg_a=*/false, a, /*neg_b=*/false, b,
        /*c_mod=*/(short)0, acc, /*reuse_a=*/false, /*reuse_b=*/false);
}

// ---------------------------------------------------------------------------
// K0: zero the reduction accumulators (must run every launch).
// ---------------------------------------------------------------------------
__global__ void zero_kernel(float* __restrict__ p, int n) {
    int i = blockIdx.x * blockDim.x + threadIdx.x;
    if (i < n) p[i] = 0.0f;
}

// ---------------------------------------------------------------------------
// K1: Wz = z @ W^T  for the two node inputs (8192 rows x 64 x 64 each).
// 128 threads = 4 waves; block handles 16 rows x 64 cols; wave w -> col tile.
// ---------------------------------------------------------------------------
__global__ __launch_bounds__(128) void node_proj_kernel(
    const float* __restrict__ z_m, const float* __restrict__ z_k,
    const float* __restrict__ Wm,  const float* __restrict__ Wk,
    float* __restrict__ Wz_m, float* __restrict__ Wz_k)
{
    __shared__ float lds_a[16 * LDA];

    const int tid  = threadIdx.x;
    const int wave = tid >> 5;
    const int lane = tid & 31;
    const int lo   = lane & 15;
    const int hi   = lane >> 4;

    const int sel = blockIdx.y;                     // 0: m-side, 1: k-side
    const float* __restrict__ src = sel ? z_k : z_m;
    const float* __restrict__ W   = sel ? Wk  : Wm;
    float* __restrict__ dst       = sel ? Wz_k : Wz_m;

    const size_t r0 = (size_t)blockIdx.x * 16;      // row tile of 8192 flat rows

    // stage A tile (16 x 64) into padded LDS via async copy
    for (int j = 0; j < 2; ++j) {
        int idx4 = tid + 128 * j;                   // float4 index 0..255
        int fl   = idx4 * 4;
        int r = fl >> 6, c = fl & 63;
        async_copy_b128(&lds_a[r * LDA + c], src + r0 * FDIM + fl);
    }
    wait_async_all();
    __syncthreads();

    const int f0 = wave * 16;
    const float* abase = &lds_a[lo * LDA + 2 * hi];
    const float* bbase = W + (size_t)(f0 + lo) * FDIM + 2 * hi;

    v8f acc = {};
#pragma unroll
    for (int s = 0; s < 16; ++s) {
        v2f a = *(const v2f*)(abase + 4 * s);
        v2f b = *(const v2f*)(bbase + 4 * s);
        acc = wmma_step(a, b, acc);
    }

#pragma unroll
    for (int i = 0; i < 8; ++i) {
        dst[(r0 + i + 8 * hi) * FDIM + f0 + lo] = acc[i];
    }
}

// ---------------------------------------------------------------------------
// K2: fused edge kernel.
// Block = 256 threads = 8 waves, owns an (8 m) x (16 k) x (64 f) tile.
// Wave w: m = m0+w, 16 consecutive rows of z_mk, all 64 output cols (4 tiles).
// Per wave:  async-stage A, GEMM (4 x 16 WMMA) + broadcast add + leaky +
//            store + in-register k-reduction (msg_nm) + LDS m-reduction
//            (msg_nk, ds_add_f32) + global f32 atomics to flush.
// ---------------------------------------------------------------------------
__global__ __launch_bounds__(256) void edge_kernel(
    const float* __restrict__ z_mk, const float* __restrict__ Wedge,
    const float* __restrict__ Wz_m, const float* __restrict__ Wz_k,
    float* __restrict__ out_mk,
    float* __restrict__ sum_nm, float* __restrict__ sum_nk)
{
    __shared__ float lds_a[8 * 16 * LDA];           // 8 waves x 16 rows x 68
    __shared__ float lds_nk[16 * FDIM];             // k x f accumulator

    const int tid  = threadIdx.x;
    const int wave = tid >> 5;
    const int lane = tid & 31;
    const int lo   = lane & 15;
    const int hi   = lane >> 4;

    const int blk = blockIdx.x;                     // b*32 + mt*4 + kt
    const int b   = blk >> 5;
    const int mt  = (blk >> 2) & 7;                 // 8 m-tiles of 8
    const int kt  = blk & 3;                        // 4 k-tiles of 16
    const int m0  = mt * 8;
    const int k0  = kt * 16;
    const int m   = m0 + wave;

    // cooperative per-wave async stage: rows (b, m, k0..k0+15) of z_mk -> LDS
    const size_t rowbase = (((size_t)b * MDIM + m) * KDIM + k0) * FDIM;
    {
        const float* srcp = z_mk + rowbase;         // 16 x 64 contiguous floats
        float* dstp = &lds_a[wave * 16 * LDA];
#pragma unroll
        for (int j = 0; j < 8; ++j) {
            int idx4 = lane + 32 * j;
            int fl   = idx4 * 4;
            int r = fl >> 6, c = fl & 63;
            async_copy_b128(dstp + r * LDA + c, srcp + fl);
        }
    }

    for (int i = tid; i < 16 * FDIM; i += 256) lds_nk[i] = 0.0f;

    wait_async_all();
    __syncthreads();

    // A fragments for the full K=64, held in registers across all col tiles
    v2f afrag[16];
    {
        const float* abase = &lds_a[wave * 16 * LDA + lo * LDA + 2 * hi];
#pragma unroll
        for (int s = 0; s < 16; ++s)
            afrag[s] = *(const v2f*)(abase + 4 * s);
    }

    const size_t node_b = (size_t)b * MDIM * FDIM;  // base for (b, node, f) arrays

    for (int ft = 0; ft < 4; ++ft) {
        const int f0 = ft * 16;
        const float* bbase = Wedge + (size_t)(f0 + lo) * FDIM + 2 * hi;

        v8f acc = {};
#pragma unroll
        for (int s = 0; s < 16; ++s) {
            v2f bf = *(const v2f*)(bbase + 4 * s);
            acc = wmma_step(afrag[s], bf, acc);
        }

        const int n = f0 + lo;
        const float wzm = Wz_m[node_b + (size_t)m * FDIM + n];
        float nm_acc = 0.0f;
#pragma unroll
        for (int i = 0; i < 8; ++i) {
            const int kl = i + 8 * hi;              // local k 0..15
            const int k  = k0 + kl;
            float v = acc[i] + wzm + Wz_k[node_b + (size_t)k * FDIM + n];
            v = leaky(v);
            out_mk[rowbase + (size_t)kl * FDIM + n] = v;
            nm_acc += v;                                     // reduce over k
            atomic_add_f32(&lds_nk[kl * FDIM + n], v);       // reduce over m
        }
        // fold lane l with l+16 (same m, same n), then one atomic per column
        nm_acc += __shfl_xor(nm_acc, 16, 32);
        if (hi == 0)
            atomic_add_f32(&sum_nm[node_b + (size_t)m * FDIM + n], nm_acc);
    }

    __syncthreads();
    for (int i = tid; i < 16 * FDIM; i += 256) {
        const int kl = i >> 6, f = i & 63;
        atomic_add_f32(&sum_nk[node_b + (size_t)(k0 + kl) * FDIM + f], lds_nk[i]);
    }
}

// ---------------------------------------------------------------------------
// K3: node update.  out = leaky(z @ Wself^T + (sum/64) @ Wneigh^T)
// Same block shape as K1, two accumulated WMMA GEMMs (32 steps).
// ---------------------------------------------------------------------------
__global__ __launch_bounds__(128) void node_update_kernel(
    const float* __restrict__ z_m, const float* __restrict__ z_k,
    const float* __restrict__ sum_nm, const float* __restrict__ sum_nk,
    const float* __restrict__ Wself_m, const float* __restrict__ Wself_k,
    const float* __restrict__ Wneigh_m, const float* __restrict__ Wneigh_k,
    float* __restrict__ out_m, float* __restrict__ out_k)
{
    __shared__ float lds_a1[16 * LDA];
    __shared__ float lds_a2[16 * LDA];

    const int tid  = threadIdx.x;
    const int wave = tid >> 5;
    const int lane = tid & 31;
    const int lo   = lane & 15;
    const int hi   = lane >> 4;

    const int sel = blockIdx.y;
    const float* __restrict__ zsrc = sel ? z_k     : z_m;
    const float* __restrict__ ssrc = sel ? sum_nk  : sum_nm;
    const float* __restrict__ Ws   = sel ? Wself_k : Wself_m;
    const float* __restrict__ Wn   = sel ? Wneigh_k : Wneigh_m;
    float* __restrict__ dst        = sel ? out_k   : out_m;

    const size_t r0 = (size_t)blockIdx.x * 16;
    const float inv = 1.0f / 64.0f;                 // mean over 64 counterparts

    for (int j = 0; j < 2; ++j) {
        int idx4 = tid + 128 * j;
        int fl   = idx4 * 4;
        int r = fl >> 6, c = fl & 63;
        v4f v1 = *(const v4f*)(zsrc + r0 * FDIM + fl);
        v4f v2 = *(const v4f*)(ssrc + r0 * FDIM + fl);
        *(v4f*)(&lds_a1[r * LDA + c]) = v1;
        v4f v2s; v2s.x = v2.x * inv; v2s.y = v2.y * inv;
        v2s.z = v2.z * inv; v2s.w = v2.w * inv;
        *(v4f*)(&lds_a2[r * LDA + c]) = v2s;
    }
    __syncthreads();

    const int f0 = wave * 16;
    const float* a1b = &lds_a1[lo * LDA + 2 * hi];
    const float* a2b = &lds_a2[lo * LDA + 2 * hi];
    const float* b1b = Ws + (size_t)(f0 + lo) * FDIM + 2 * hi;
    const float* b2b = Wn + (size_t)(f0 + lo) * FDIM + 2 * hi;

    v8f acc = {};
#pragma unroll
    for (int s = 0; s < 16; ++s)
        acc = wmma_step(*(const v2f*)(a1b + 4 * s), *(const v2f*)(b1b + 4 * s), acc);
#pragma unroll
    for (int s = 0; s < 16; ++s)
        acc = wmma_step(*(const v2f*)(a2b + 4 * s), *(const v2f*)(b2b + 4 * s), acc);

#pragma unroll
    for (int i = 0; i < 8; ++i)
        dst[(r0 + i + 8 * hi) * FDIM + f0 + lo] = leaky(acc[i]);
}

// ---------------------------------------------------------------------------
extern "C" void kernel_launch(void* const* d_in, const int* in_sizes, int n_in,
                              void* d_out, int out_size, void* d_ws, size_t ws_size,
                              hipStream_t stream) {
    const float* z_mk     = (const float*)d_in[0];
    const float* z_m      = (const float*)d_in[1];
    const float* z_k      = (const float*)d_in[2];
    const float* Wedge    = (const float*)d_in[3];
    const float* Wm       = (const float*)d_in[4];
    const float* Wk       = (const float*)d_in[5];
    const float* Wself_m  = (const float*)d_in[6];
    const float* Wself_k  = (const float*)d_in[7];
    const float* Wneigh_m = (const float*)d_in[8];
    const float* Wneigh_k = (const float*)d_in[9];

    float* out    = (float*)d_out;
    float* out_mk = out;
    float* out_m  = out + (size_t)N_MK;
    float* out_k  = out_m + (size_t)N_NODE;

    float* ws     = (float*)d_ws;
    float* Wz_m   = ws;
    float* Wz_k   = ws + 1 * (size_t)N_NODE;
    float* sum_nm = ws + 2 * (size_t)N_NODE;
    float* sum_nk = ws + 3 * (size_t)N_NODE;

    // K0: clear accumulators (sum_nm, sum_nk are contiguous)
    zero_kernel<<<(2 * N_NODE + 255) / 256, 256, 0, stream>>>(sum_nm, 2 * N_NODE);

    // K1: node projections
    node_proj_kernel<<<dim3(512, 2), 128, 0, stream>>>(z_m, z_k, Wm, Wk, Wz_m, Wz_k);

    // K2: fused edge GEMM + activation + both mean reductions
    edge_kernel<<<BATCH * 32, 256, 0, stream>>>(z_mk, Wedge, Wz_m, Wz_k,
                                                out_mk, sum_nm, sum_nk);

    // K3: node updates
    node_update_kernel<<<dim3(512, 2), 128, 0, stream>>>(
        z_m, z_k, sum_nm, sum_nk, Wself_m, Wself_k, Wneigh_m, Wneigh_k,
        out_m, out_k);
}